// GineBlock_44564580663332
// MI455X (gfx1250) — compile-verified
//
#include <hip/hip_runtime.h>

#define NODE_DIM 128
#define NNODES   50000
#define NEDGES   600000
#define BN_EPS   1e-5f
#define LDS_PITCH 130   // 64-bank conflict avoidance for column-pair reads

typedef float v2f __attribute__((ext_vector_type(2)));
typedef float v8f __attribute__((ext_vector_type(8)));

// ---------------------------------------------------------------- zero scratch
__global__ void k_zero(float* __restrict__ agg, float* __restrict__ sums) {
    const size_t NK = (size_t)NNODES * NODE_DIM;
    size_t i = (size_t)blockIdx.x * blockDim.x + threadIdx.x;
    if (i < NK)            agg[i] = 0.0f;
    else if (i < NK + 256) sums[i - NK] = 0.0f;
}

// ------------------------------------------------- edge message + scatter add
// 32 lanes per edge, 4 features per lane (float4 loads, f32 global atomics).
__global__ void k_edge(const float* __restrict__ x,
                       const int*   __restrict__ ei,
                       const float* __restrict__ ea,
                       float*       __restrict__ agg) {
    unsigned gid = blockIdx.x * blockDim.x + threadIdx.x;
    unsigned e = gid >> 5;
    if (e >= NEDGES) return;
    unsigned c = (gid & 31u) << 2;
    int s = ei[e];
    int d = ei[NEDGES + e];
    const float4 xv = *(const float4*)(x  + (size_t)s * NODE_DIM + c);
    const float4 ev = *(const float4*)(ea + (size_t)e * NODE_DIM + c);
    float* ag = agg + (size_t)d * NODE_DIM + c;
    atomicAdd(ag + 0, fmaxf(xv.x + ev.x, 0.0f));
    atomicAdd(ag + 1, fmaxf(xv.y + ev.y, 0.0f));
    atomicAdd(ag + 2, fmaxf(xv.z + ev.z, 0.0f));
    atomicAdd(ag + 3, fmaxf(xv.w + ev.w, 0.0f));
}

// --------------------------- fused (x+agg) -> GEMM1 -> ReLU -> GEMM2 -> stats
// 128 threads = 4 waves; block owns 64 rows; wave w owns rows 16w..16w+15.
__global__ __launch_bounds__(128) void k_gemm(
        const float* __restrict__ x,   const float* __restrict__ agg,
        const float* __restrict__ w1,  const float* __restrict__ b1,
        const float* __restrict__ w2,  const float* __restrict__ b2,
        float* __restrict__ y2,
        float* __restrict__ gsum,      float* __restrict__ gsumsq) {
    __shared__ float lds[64 * LDS_PITCH];

    const int tid  = threadIdx.x;
    const int wave = tid >> 5;
    const int lane = tid & 31;
    const int m    = lane & 15;    // row/col within 16
    const int hl   = lane >> 4;    // lane half selects K pair / M half
    const long long m0 = (long long)blockIdx.x * 64;

    // stage h0 = x + agg (coalesced, one row of 128 per iteration)
    for (int r = 0; r < 64; ++r) {
        long long row = m0 + r;
        float v = 0.0f;
        if (row < NNODES) {
            size_t idx = (size_t)row * NODE_DIM + tid;
            v = x[idx] + agg[idx];
        }
        lds[r * LDS_PITCH + tid] = v;
    }
    __syncthreads();

    const int arow = wave * 16 + m;

    // ---- GEMM1: y1 = relu(h0 @ w1.T + b1)
    v8f acc[8] = {};
    for (int ks = 0; ks < 32; ++ks) {
        const int kb = ks * 4 + hl * 2;
        v2f a;
        a.x = lds[arow * LDS_PITCH + kb];
        a.y = lds[arow * LDS_PITCH + kb + 1];
#pragma unroll
        for (int nt = 0; nt < 8; ++nt) {
            const int n = nt * 16 + m;
            v2f b;
            b.x = w1[n * NODE_DIM + kb];      // B[k][n] = w1[n][k]
            b.y = w1[n * NODE_DIM + kb + 1];
            acc[nt] = __builtin_amdgcn_wmma_f32_16x16x4_f32(
                false, a, false, b, (short)0, acc[nt], false, false);
        }
    }
    __syncthreads();   // everyone done reading h0 before overwrite

    // write y1 (bias + relu) back into the same LDS buffer
#pragma unroll
    for (int nt = 0; nt < 8; ++nt) {
        const int n = nt * 16 + m;
        const float bias = b1[n];
#pragma unroll
        for (int j = 0; j < 8; ++j) {
            const int rr = wave * 16 + j + hl * 8;   // D layout: M = j + 8*half
            lds[rr * LDS_PITCH + n] = fmaxf(acc[nt][j] + bias, 0.0f);
        }
    }
    __syncthreads();

    // ---- GEMM2: y2 = y1 @ w2.T + b2
#pragma unroll
    for (int nt = 0; nt < 8; ++nt) acc[nt] = (v8f)0.0f;
    for (int ks = 0; ks < 32; ++ks) {
        const int kb = ks * 4 + hl * 2;
        v2f a;
        a.x = lds[arow * LDS_PITCH + kb];
        a.y = lds[arow * LDS_PITCH + kb + 1];
#pragma unroll
        for (int nt = 0; nt < 8; ++nt) {
            const int n = nt * 16 + m;
            v2f b;
            b.x = w2[n * NODE_DIM + kb];
            b.y = w2[n * NODE_DIM + kb + 1];
            acc[nt] = __builtin_amdgcn_wmma_f32_16x16x4_f32(
                false, a, false, b, (short)0, acc[nt], false, false);
        }
    }

    // epilogue: bias, store y2, per-lane BN column partial sums
    float cs[8], cq[8];
#pragma unroll
    for (int nt = 0; nt < 8; ++nt) {
        const int n = nt * 16 + m;
        const float bias = b2[n];
        float s = 0.0f, q = 0.0f;
#pragma unroll
        for (int j = 0; j < 8; ++j) {
            const int rl = wave * 16 + j + hl * 8;
            const long long row = m0 + rl;
            if (row < NNODES) {
                const float v = acc[nt][j] + bias;
                y2[(size_t)row * NODE_DIM + n] = v;
                s += v;
                q += v * v;
            }
        }
        cs[nt] = s;
        cq[nt] = q;
    }

    // block-level column reduction in LDS, then one global atomic per column
    __syncthreads();
    lds[tid]       = 0.0f;
    lds[128 + tid] = 0.0f;
    __syncthreads();
#pragma unroll
    for (int nt = 0; nt < 8; ++nt) {
        const int n = nt * 16 + m;
        atomicAdd(&lds[n],       cs[nt]);
        atomicAdd(&lds[128 + n], cq[nt]);
    }
    __syncthreads();
    atomicAdd(&gsum[tid],   lds[tid]);
    atomicAdd(&gsumsq[tid], lds[128 + tid]);
}

// -------------------------------------------------------- batchnorm + relu
__global__ void k_bn(const float* __restrict__ y2,
                     const float* __restrict__ gsum,
                     const float* __restrict__ gsumsq,
                     const float* __restrict__ bw,
                     const float* __restrict__ bb,
                     float* __restrict__ out) {
    size_t i = (size_t)blockIdx.x * blockDim.x + threadIdx.x;
    if (i >= (size_t)NNODES * NODE_DIM) return;
    const int f = (int)(i & (NODE_DIM - 1));
    const float invN = 1.0f / (float)NNODES;
    const float mean = gsum[f] * invN;
    const float var  = gsumsq[f] * invN - mean * mean;
    const float sc   = rsqrtf(var + BN_EPS) * bw[f];
    const float v    = (y2[i] - mean) * sc + bb[f];
    out[i] = fmaxf(v, 0.0f);
}

extern "C" void kernel_launch(void* const* d_in, const int* in_sizes, int n_in,
                              void* d_out, int out_size, void* d_ws, size_t ws_size,
                              hipStream_t stream) {
    (void)in_sizes; (void)n_in; (void)out_size; (void)ws_size;
    const float* x  = (const float*)d_in[0];
    const int*   ei = (const int*)  d_in[1];
    const float* ea = (const float*)d_in[2];
    const float* w1 = (const float*)d_in[3];
    const float* b1 = (const float*)d_in[4];
    const float* w2 = (const float*)d_in[5];
    const float* b2 = (const float*)d_in[6];
    const float* bw = (const float*)d_in[7];
    const float* bb = (const float*)d_in[8];
    float* out = (float*)d_out;

    const size_t NK = (size_t)NNODES * NODE_DIM;
    float* agg    = (float*)d_ws;          // [N,128]
    float* y2     = agg + NK;              // [N,128]
    float* gsum   = y2 + NK;               // [128]
    float* gsumsq = gsum + NODE_DIM;       // [128]

    const size_t nz = NK + 256;
    k_zero<<<(unsigned)((nz + 255) / 256), 256, 0, stream>>>(agg, gsum);

    const size_t ethreads = (size_t)NEDGES * 32;
    k_edge<<<(unsigned)((ethreads + 255) / 256), 256, 0, stream>>>(x, ei, ea, agg);

    k_gemm<<<(NNODES + 63) / 64, 128, 0, stream>>>(x, agg, w1, b1, w2, b2,
                                                   y2, gsum, gsumsq);

    k_bn<<<(unsigned)((NK + 255) / 256), 256, 0, stream>>>(y2, gsum, gsumsq,
                                                           bw, bb, out);
}